// CenterLoss_76115410420300
// MI455X (gfx1250) — compile-verified
//
#include <hip/hip_runtime.h>
#include <math.h>

#define NCLS 1000
#define FDIM 256
#define NROW 262144
#define ALPHA 0.5f

// ws layout (floats): sums[NCLS*FDIM], counts[NCLS], loss[1]
#define WS_SUMS 0
#define WS_CNT  (NCLS * FDIM)
#define WS_LOSS (NCLS * FDIM + NCLS)
#define WS_TOTAL (NCLS * FDIM + NCLS + 1)

// segment-sum pass geometry
#define CHUNK 32                       // columns per workgroup
#define NCHUNK (FDIM / CHUNK)          // 8
#define RGROUPS 32                     // row groups
#define ROWS_PER_WG (NROW / RGROUPS)   // 8192
#define WAVES_PER_WG 8
#define ROWS_PER_WAVE (ROWS_PER_WG / WAVES_PER_WG)  // 1024
#define TROWS 16                       // rows per TDM tile
#define NTILES (ROWS_PER_WAVE / TROWS) // 64
#define TILE_FLOATS (TROWS * CHUNK)    // 512 floats = 2048 B per tile buffer

typedef __attribute__((ext_vector_type(4))) unsigned int v4u;
typedef __attribute__((ext_vector_type(4))) int v4i;
typedef __attribute__((ext_vector_type(8))) int v8i;

// ---------------------------------------------------------------------------
// CDNA5 Tensor Data Mover: one 2D strided tile, global -> LDS (TENSORcnt).
// D# group0 = {flags(count=1), lds_addr, global_addr_lo,
//              global_addr_hi | type=2<<30}; group1 = dims/strides; groups
// 2/3 zero (2-D tile). Builtin arity differs per toolchain.
// ---------------------------------------------------------------------------
__device__ __forceinline__ void tdm_load_tile(v4u g0, v8i g1) {
  v4i z4 = (v4i)0;
#if __clang_major__ >= 23
  v8i z8 = (v8i)0;
  __builtin_amdgcn_tensor_load_to_lds(g0, g1, z4, z4, z8, 0);
#else
  __builtin_amdgcn_tensor_load_to_lds(g0, g1, z4, z4, 0);
#endif
}

// CDNA5 async global->LDS (ASYNCcnt): vdst = LDS byte offset, vaddr = 64-bit.
__device__ __forceinline__ void async_b128_to_lds(unsigned lds_byte_off,
                                                  unsigned long long gaddr) {
  asm volatile("global_load_async_to_lds_b128 %0, %1, off"
               :: "v"(lds_byte_off), "v"(gaddr)
               : "memory");
}
__device__ __forceinline__ void wait_async_le1() {
  asm volatile("s_wait_asynccnt 1" ::: "memory");
}
__device__ __forceinline__ void wait_async_le0() {
  asm volatile("s_wait_asynccnt 0" ::: "memory");
}

// ---------------------------------------------------------------------------
__global__ void zero_ws_kernel(float* __restrict__ ws) {
  for (int i = blockIdx.x * blockDim.x + threadIdx.x; i < WS_TOTAL;
       i += gridDim.x * blockDim.x)
    ws[i] = 0.0f;
}

// ---------------------------------------------------------------------------
// Pass 1: loss = mean(sqrt(sum_d (x - centers[label])^2)) / 2, plus counts.
// ---------------------------------------------------------------------------
__global__ __launch_bounds__(256) void loss_kernel(
    const float* __restrict__ x, const int* __restrict__ labels,
    const float* __restrict__ centers, float* __restrict__ ws) {
  const int lane = threadIdx.x & 31;
  const int wave = threadIdx.x >> 5;
  const int wgid = blockIdx.x * (blockDim.x >> 5) + wave;  // global wave id
  const long long rbase = (long long)wgid * 64;

  const float4* __restrict__ x4 = (const float4*)x;
  const float4* __restrict__ c4 = (const float4*)centers;

  float acc = 0.0f;
  for (int ch = 0; ch < 2; ++ch) {
    const long long cb = rbase + ch * 32;
    int labv = labels[cb + lane];                 // label for row cb+lane
    atomicAdd(&ws[WS_CNT + labv], 1.0f);          // per-class counts

    for (int r = 0; r < 32; ++r) {
      const long long row = cb + r;
      const int lab = __shfl(labv, r, 32);
      // CDNA5 prefetch (global_prefetch_b8) for the row 2 ahead
      __builtin_prefetch(&x4[(size_t)(row + 2) * (FDIM / 4) + lane * 2], 0, 1);

      float4 a0 = x4[(size_t)row * (FDIM / 4) + lane * 2];
      float4 a1 = x4[(size_t)row * (FDIM / 4) + lane * 2 + 1];
      float4 b0 = c4[(size_t)lab * (FDIM / 4) + lane * 2];
      float4 b1 = c4[(size_t)lab * (FDIM / 4) + lane * 2 + 1];

      float d0 = a0.x - b0.x, d1 = a0.y - b0.y, d2 = a0.z - b0.z, d3 = a0.w - b0.w;
      float d4 = a1.x - b1.x, d5 = a1.y - b1.y, d6 = a1.z - b1.z, d7 = a1.w - b1.w;
      float p = d0 * d0 + d1 * d1 + d2 * d2 + d3 * d3 +
                d4 * d4 + d5 * d5 + d6 * d6 + d7 * d7;
      // wave32 butterfly reduce
      p += __shfl_xor(p, 16, 32);
      p += __shfl_xor(p, 8, 32);
      p += __shfl_xor(p, 4, 32);
      p += __shfl_xor(p, 2, 32);
      p += __shfl_xor(p, 1, 32);
      acc += sqrtf(p);
    }
  }
  if (lane == 0) atomicAdd(&ws[WS_LOSS], acc);
}

// ---------------------------------------------------------------------------
// Pass 2: segment sums. Grid = (NCHUNK, RGROUPS). Each WG owns a 32-column
// chunk for 8192 rows. Per wave/tile: one TDM tensor_load_to_lds moves the
// 16x128B x-tile (row stride 1KB), one async b128 op (lanes 0-3) stages the
// 16 labels; double-buffered on TENSORcnt/ASYNCcnt. Accumulation is
// ds_add_f32 into a 1000x32 LDS array; one atomic flush to global at the end.
// ---------------------------------------------------------------------------
__global__ __launch_bounds__(256) void seg_kernel(
    const float* __restrict__ x, const int* __restrict__ labels,
    float* __restrict__ ws) {
  __shared__ float lsum[NCLS * CHUNK];                                  // 128000 B
  __shared__ __align__(16) float stage[WAVES_PER_WG * 2 * TILE_FLOATS]; // 32768 B
  __shared__ __align__(16) int lstage[WAVES_PER_WG * 2 * TROWS];        // 1024 B

  const int chunk = blockIdx.x;      // 0..7
  const int rg = blockIdx.y;         // 0..31
  const int lane = threadIdx.x & 31;
  const int wave = threadIdx.x >> 5;
  const int c0 = chunk * CHUNK;

  for (int i = threadIdx.x; i < NCLS * CHUNK; i += 256) lsum[i] = 0.0f;
  __syncthreads();

  const int r0w = rg * ROWS_PER_WG + wave * ROWS_PER_WAVE;
  float* wstage = &stage[wave * (2 * TILE_FLOATS)];
  int* wlab = &lstage[wave * (2 * TROWS)];
  const unsigned wstage_off = (unsigned)(size_t)wstage;  // LDS byte offsets
  const unsigned wlab_off = (unsigned)(size_t)wlab;
  const char* xg = (const char*)x;
  const char* lg = (const char*)labels;

  // Constant D# group1: wg_mask=0, data_size=4B (code 2), tensor_dim0=FDIM,
  // tensor_dim1=NROW, tile 32x16, tensor_dim0_stride=FDIM elements.
  const v8i g1 = { 0x20000,
                   (int)(FDIM << 16),
                   0,
                   (int)((NROW >> 16) | (CHUNK << 16)),
                   TROWS,
                   FDIM,
                   0, 0 };

  auto issue = [&](int t, int b) {
    const int rt = r0w + t * TROWS;
    // TDM: 2D strided x-tile (16 rows x 128 B) global -> LDS
    unsigned long long ga =
        (unsigned long long)(xg + (((size_t)rt * FDIM + c0) << 2));
    v4u g0 = { 1u, wstage_off + (unsigned)(b * (TILE_FLOATS * 4)),
               (unsigned)ga, (unsigned)(ga >> 32) | 0x80000000u };
    tdm_load_tile(g0, g1);
    // async global->LDS: this tile's 16 labels (lanes 0..3 x b128)
    if (lane < 4) {
      unsigned lds = wlab_off + (unsigned)(b * (TROWS * 4)) + (lane << 4);
      unsigned long long gl =
          (unsigned long long)(lg + ((size_t)rt << 2) + (lane << 4));
      async_b128_to_lds(lds, gl);
    }
  };

  issue(0, 0);
  for (int t = 0; t < NTILES; ++t) {
    const int b = t & 1;
    if (t + 1 < NTILES) {
      issue(t + 1, b ^ 1);
      __builtin_amdgcn_s_wait_tensorcnt((short)1);  // tile t's TDM op done
      wait_async_le1();                             // tile t's label op done
    } else {
      __builtin_amdgcn_s_wait_tensorcnt((short)0);
      wait_async_le0();
    }
    const float* buf = wstage + b * TILE_FLOATS;
    const int* lbuf = wlab + b * TROWS;
    for (int r = 0; r < TROWS; ++r) {
      const int lab = lbuf[r];                     // uniform ds_load broadcast
      float v = buf[r * CHUNK + lane];
      atomicAdd(&lsum[lab * CHUNK + lane], v);     // ds_add_f32, conflict-free
    }
  }
  __syncthreads();

  // flush aggregated partials: 32000 global atomics per WG (8x fewer than naive)
  for (int i = threadIdx.x; i < NCLS * CHUNK; i += 256) {
    const int cls = i / CHUNK;
    const int d = i % CHUNK;
    float v = lsum[i];
    if (v != 0.0f) atomicAdd(&ws[WS_SUMS + cls * FDIM + c0 + d], v);
  }
}

// ---------------------------------------------------------------------------
// Pass 3: finalize. d_out[0] = loss, d_out[1..] = new_centers.
// ---------------------------------------------------------------------------
__global__ void finalize_kernel(const float* __restrict__ centers,
                                const float* __restrict__ ws,
                                float* __restrict__ out) {
  const int i = blockIdx.x * blockDim.x + threadIdx.x;
  if (i == 0) out[0] = ws[WS_LOSS] * (0.5f / (float)NROW);
  if (i < NCLS * FDIM) {
    const int cls = i / FDIM;
    const float cnt = ws[WS_CNT + cls];
    const float c = centers[i];
    const float mean = ws[WS_SUMS + i] / fmaxf(cnt, 1.0f);
    out[1 + i] = (cnt > 0.0f) ? (c + ALPHA * (mean - c)) : c;
  }
}

// ---------------------------------------------------------------------------
extern "C" void kernel_launch(void* const* d_in, const int* in_sizes, int n_in,
                              void* d_out, int out_size, void* d_ws,
                              size_t ws_size, hipStream_t stream) {
  (void)in_sizes; (void)n_in; (void)out_size; (void)ws_size;
  const float* x = (const float*)d_in[0];
  const int* labels = (const int*)d_in[1];
  const float* centers = (const float*)d_in[2];
  float* out = (float*)d_out;
  float* ws = (float*)d_ws;

  zero_ws_kernel<<<512, 256, 0, stream>>>(ws);
  loss_kernel<<<512, 256, 0, stream>>>(x, labels, centers, ws);  // 4096 waves x 64 rows
  seg_kernel<<<dim3(NCHUNK, RGROUPS), 256, 0, stream>>>(x, labels, ws);
  finalize_kernel<<<(NCLS * FDIM + 255) / 256, 256, 0, stream>>>(centers, ws, out);
}